// _DANetHead_25194278158898
// MI455X (gfx1250) — compile-verified
//
#include <hip/hip_runtime.h>
#include <stdint.h>

// ---------------------------------------------------------------------------
// DANet head for MI455X (gfx1250, wave32).
//  - All matmuls on V_WMMA_F32_16X16X4_F32 (full fp32 fidelity; workload is
//    HBM-bound: ~0.87 TFLOP vs ~2 GB traffic @ 23.3 TB/s, fp32 pipe is free).
//  - GEMM tile staging uses the Tensor Data Mover (TENSOR_LOAD_TO_LDS, one
//    instruction per 2-D strided tile, hardware-inserted LDS padding that
//    matches the bank-conflict-free pitches), tracked with TENSORcnt.
//  - Conv weight staging uses GLOBAL_LOAD_ASYNC_TO_LDS_B128 (ASYNCcnt).
//  - LDS is double-buffered: DMA of chunk i+1 overlaps WMMA of chunk i.
// ---------------------------------------------------------------------------

typedef __attribute__((ext_vector_type(2))) float v2f;
typedef __attribute__((ext_vector_type(8))) float v8f;
typedef uint32_t u32x4 __attribute__((ext_vector_type(4)));
typedef uint32_t u32x8 __attribute__((ext_vector_type(8)));

__device__ __forceinline__ v8f wmma_f32(v2f a, v2f b, v8f c) {
  return __builtin_amdgcn_wmma_f32_16x16x4_f32(false, a, false, b, (short)0, c,
                                               false, false);
}

// Async DMA: global -> LDS, 16B per lane (512B per wave-instruction).
// Low 32 bits of a generic pointer to __shared__ == wave-relative LDS address.
__device__ __forceinline__ void async_load_b128(void* lds, const float* g) {
  uint32_t l = (uint32_t)(uintptr_t)lds;
  asm volatile("global_load_async_to_lds_b128 %0, %1, off"
               :
               : "v"(l), "v"(g)
               : "memory");
}
__device__ __forceinline__ void wait_asynccnt0() {
  asm volatile("s_wait_asynccnt 0x0" ::: "memory");
}

// TDM: one instruction DMAs a 2-D tile (tile_w x tile_h f32, row stride
// `stride` elems) from global into LDS at `lds_addr`, inserting `pad_amt`+1
// DWORDs of padding after every 2^(pad_int+1) DWORDs (i.e. after each row).
// Descriptor per CDNA5 ISA 8.3/8.4 (D# groups 0/1; groups 2/3 NULL for 2-D).
__device__ __forceinline__ void tdm_load_2d(uint32_t lds_addr, const float* g,
                                            uint32_t tile_w, uint32_t tile_h,
                                            uint32_t stride, uint32_t pad_int,
                                            uint32_t pad_amt) {
  uint64_t ga = (uint64_t)(uintptr_t)g;
  u32x4 g0;
  g0[0] = 1u;                                  // count=1 (valid), user mode
  g0[1] = lds_addr;                            // lds_addr [63:32]
  g0[2] = (uint32_t)ga;                        // global_addr [95:64]
  g0[3] = (uint32_t)(ga >> 32) | (2u << 30);   // global_addr hi, type=2
  u32x8 g1;
  g1[0] = (2u << 16) | (1u << 20) | (pad_int << 22) | (pad_amt << 25);
  g1[1] = 0xFFFF0000u;                // tensor_dim0 = 0x7FFFFFFF (lo16 << 16)
  g1[2] = 0xFFFF7FFFu;                // tensor_dim0 hi16 | tensor_dim1 lo16
  g1[3] = 0x7FFFu | (tile_w << 16);   // tensor_dim1 hi16 | tile_dim0
  g1[4] = tile_h;                     // tile_dim1 | tile_dim2=0
  g1[5] = stride;                     // tensor_dim0_stride lo32
  g1[6] = 0u;                         // stride hi16 | tensor_dim1_stride lo16
  g1[7] = 0u;
  asm volatile("tensor_load_to_lds %0, %1" : : "s"(g0), "s"(g1) : "memory");
}
__device__ __forceinline__ void wait_tensorcnt0() {
  asm volatile("s_wait_tensorcnt 0x0" ::: "memory");
}

#define BM 64
#define BN 128
#define BK 16
#define LDA 20   // As row pitch (floats): conflict-free fragment reads
#define LDB 136  // Bs row pitch (floats): lane halves hit disjoint bank sets

// ===========================================================================
// Generic tiled GEMM:  C[M,N] = A x B   (+ epilogue)
//   AT=0: A stored [M][K] ; AT=1: A stored [K][M]
//   BT=0: B stored [K][N] ; BT=1: B stored [N][K]
//   EPI=0: C=acc   EPI=1: C=acc+bias[m]   EPI=2: C=gamma[0]*acc+Res[m][n]
//   MASK=1: guard A loads/stores with m<M (conv8, M=19) -> VGPR staging only
// Requirements: K%16==0, N%128==0. TDM staging used when rows are full.
// ===========================================================================
template <int AT, int BT, int EPI, int MASK>
__global__ __launch_bounds__(256) void gemm_wmma_k(
    const float* __restrict__ A, const float* __restrict__ B,
    float* __restrict__ C, int M, int N, int K, long long sA, long long sB,
    long long sC, const float* __restrict__ P, const float* __restrict__ R,
    long long sR) {
  __shared__ alignas(16) float As[2][BM][LDA];
  __shared__ alignas(16) float Bs[2][BK][LDB];
  const int tid = threadIdx.x;
  const int lane = tid & 31;
  const int wave = tid >> 5;
  const int wm = wave & 3;   // 4 waves along M (16 rows each)
  const int wn = wave >> 2;  // 2 waves along N (64 cols each)
  const int bz = blockIdx.z;
  A += (long long)bz * sA;
  B += (long long)bz * sB;
  C += (long long)bz * sC;
  const float* Rb = R + (long long)bz * sR;

  const int m0 = blockIdx.y * BM;
  const int n0 = blockIdx.x * BN;

  constexpr bool TDM_A = (AT == 0) && (MASK == 0);
  constexpr bool TDM_B = (BT == 0) && (MASK == 0);

  v8f acc[4];
  v8f zero = {};
#pragma unroll
  for (int t = 0; t < 4; ++t) acc[t] = zero;

  auto stage = [&](int bu, int k0) {
    // ---- TDM paths: single descriptor per tile, issued by wave 0 ----
    if (TDM_A || TDM_B) {
      if (wave == 0) {
        if (TDM_A)  // 64 rows x 16 floats, row stride K; LDS pitch 20 (pad 4)
          tdm_load_2d((uint32_t)(uintptr_t)&As[bu][0][0],
                      A + (long long)m0 * K + k0, 16u, 64u, (uint32_t)K, 3u,
                      3u);
        if (TDM_B)  // 16 rows x 128 floats, row stride N; LDS pitch 136 (pad 8)
          tdm_load_2d((uint32_t)(uintptr_t)&Bs[bu][0][0],
                      B + (long long)k0 * N + n0, 128u, 16u, (uint32_t)N, 6u,
                      7u);
      }
    }
    // ---- VGPR fallback paths ----
    if (!TDM_A) {
      if (AT == 0) {  // masked [M][K]
        const int m_l = tid >> 2, kq = (tid & 3) * 4;
        const int m = m0 + m_l;
        float4 va = make_float4(0.f, 0.f, 0.f, 0.f);
        if (m < M)
          va = *reinterpret_cast<const float4*>(A + (long long)m * K + k0 + kq);
        As[bu][m_l][kq + 0] = va.x;
        As[bu][m_l][kq + 1] = va.y;
        As[bu][m_l][kq + 2] = va.z;
        As[bu][m_l][kq + 3] = va.w;
      } else {  // AT==1: A stored [K][M], transposed scatter into LDS
        const int k_l = tid >> 4, mq = (tid & 15) * 4;
        float4 va = *reinterpret_cast<const float4*>(
            A + (long long)(k0 + k_l) * M + m0 + mq);
        As[bu][mq + 0][k_l] = va.x;
        As[bu][mq + 1][k_l] = va.y;
        As[bu][mq + 2][k_l] = va.z;
        As[bu][mq + 3][k_l] = va.w;
      }
    }
    if (!TDM_B) {
      if (BT == 0) {
        const int k_l = tid >> 4, nq = (tid & 15) * 8;
        const float* src = B + (long long)(k0 + k_l) * N + n0 + nq;
        float4 v0 = *reinterpret_cast<const float4*>(src);
        float4 v1 = *reinterpret_cast<const float4*>(src + 4);
        Bs[bu][k_l][nq + 0] = v0.x; Bs[bu][k_l][nq + 1] = v0.y;
        Bs[bu][k_l][nq + 2] = v0.z; Bs[bu][k_l][nq + 3] = v0.w;
        Bs[bu][k_l][nq + 4] = v1.x; Bs[bu][k_l][nq + 5] = v1.y;
        Bs[bu][k_l][nq + 6] = v1.z; Bs[bu][k_l][nq + 7] = v1.w;
      } else {  // BT==1: B stored [N][K], transposed scatter into LDS
        const int n_l = tid >> 1, kq = (tid & 1) * 8;
        const float* src = B + (long long)(n0 + n_l) * K + k0 + kq;
        float4 v0 = *reinterpret_cast<const float4*>(src);
        float4 v1 = *reinterpret_cast<const float4*>(src + 4);
        Bs[bu][kq + 0][n_l] = v0.x; Bs[bu][kq + 1][n_l] = v0.y;
        Bs[bu][kq + 2][n_l] = v0.z; Bs[bu][kq + 3][n_l] = v0.w;
        Bs[bu][kq + 4][n_l] = v1.x; Bs[bu][kq + 5][n_l] = v1.y;
        Bs[bu][kq + 6][n_l] = v1.z; Bs[bu][kq + 7][n_l] = v1.w;
      }
    }
  };

  auto commit = [&]() {
    if (TDM_A || TDM_B) wait_tensorcnt0();  // 0 on non-issuing waves: free
    __syncthreads();
  };

  auto mma_tile = [&](int bu) {
#pragma unroll
    for (int ks = 0; ks < 4; ++ks) {
      const int kk = ks * 4 + ((lane >> 4) << 1);  // lanes 0-15: K0/1, 16-31: K2/3
      const int ml = wm * 16 + (lane & 15);
      v2f af;
      af.x = As[bu][ml][kk];
      af.y = As[bu][ml][kk + 1];
#pragma unroll
      for (int t = 0; t < 4; ++t) {
        const int nl = wn * 64 + t * 16 + (lane & 15);
        v2f bf;
        bf.x = Bs[bu][kk][nl];
        bf.y = Bs[bu][kk + 1][nl];
        acc[t] = wmma_f32(af, bf, acc[t]);
      }
    }
  };

  // ---- double-buffered main loop ----
  stage(0, 0);
  commit();
  int buf = 0;
  for (int k0 = 0; k0 < K; k0 += BK) {
    if (k0 + BK < K) stage(buf ^ 1, k0 + BK);
    mma_tile(buf);
    commit();
    buf ^= 1;
  }

  // ---- epilogue; C/D layout: VGPR r -> M=r (lanes<16) / r+8 (lanes>=16) ----
  const int half = lane >> 4, col = lane & 15;
  float g = 0.0f;
  if (EPI == 2) g = P[0];
#pragma unroll
  for (int r = 0; r < 8; ++r) {
    const int m = m0 + wm * 16 + r + 8 * half;
    if (m < M) {
      float bias = 0.0f;
      if (EPI == 1) bias = P[m];
#pragma unroll
      for (int t = 0; t < 4; ++t) {
        const int n = n0 + wn * 64 + t * 16 + col;
        float v = acc[t][r];
        if (EPI == 1) v += bias;
        if (EPI == 2) v = g * v + Rb[(long long)m * N + n];
        C[(long long)m * N + n] = v;
      }
    }
  }
}

// ===========================================================================
// Implicit-GEMM 3x3 SAME conv + eval-BatchNorm + ReLU.
//   M=512 (Cout), K=Cin*9, N=4096 (64x64).  grid=(N/128, 512/64, B)
// Weights staged via async LDS-DMA; zero-padded input gather via VGPR path.
// ===========================================================================
__global__ __launch_bounds__(256) void conv3x3_bnrelu_k(
    const float* __restrict__ X, const float* __restrict__ Wt,
    const float* __restrict__ gs, const float* __restrict__ gb,
    const float* __restrict__ gm, const float* __restrict__ gv,
    float* __restrict__ Out, int Cin) {
  __shared__ alignas(16) float As[2][BM][LDA];
  __shared__ alignas(16) float Bs[2][BK][LDB];
  const int tid = threadIdx.x;
  const int lane = tid & 31;
  const int wave = tid >> 5;
  const int wm = wave & 3;
  const int wn = wave >> 2;
  const int bz = blockIdx.z;
  const int K = Cin * 9;
  const float* Xb = X + (long long)bz * Cin * 4096;
  float* Ob = Out + (long long)bz * 512 * 4096;
  const int m0 = blockIdx.y * BM;
  const int n0 = blockIdx.x * BN;

  v8f acc[4];
  v8f zero = {};
#pragma unroll
  for (int t = 0; t < 4; ++t) acc[t] = zero;

  const int arow = wave * 8 + (lane >> 2);  // A async: 8 rows/wave
  const int acq = (lane & 3) * 4;
  const int bk_l = tid >> 4, bnq = (tid & 15) * 8;  // B gather: 8 px/thread
  const int y0 = (n0 + bnq) >> 6;  // K-invariant pixel decomposition
  const int x0 = (n0 + bnq) & 63;  // bnq%8==0 -> no row crossing in 8 px

  // incremental (cin, tap) for k = k0 + bk_l; avoids /9 in the loop
  int cin = bk_l / 9;
  int tap = bk_l - cin * 9;

  auto stage = [&](int bu, int k0) {
    async_load_b128(&As[bu][arow][acq],
                    Wt + (long long)(m0 + arow) * K + k0 + acq);
    const int d3 = tap / 3;
    const int dy = d3 - 1;
    const int dx = tap - d3 * 3 - 1;
    const float* Xc = Xb + (long long)cin * 4096;
    const int yb = y0 + dy;
    const int rowoff = yb * 64 + x0 + dx;
    const bool yok = (unsigned)yb < 64u;
    if (cin + 1 < Cin)  // prefetch next input channel's row
      __builtin_prefetch(Xc + 4096 + (y0 * 64 + x0), 0, 0);
#pragma unroll
    for (int j = 0; j < 8; ++j) {
      const int x = x0 + dx + j;
      float v = 0.0f;
      if (yok && (unsigned)x < 64u) v = Xc[rowoff + j];
      Bs[bu][bk_l][bnq + j] = v;
    }
    // advance (cin, tap) by BK=16: 16 = 9 + 7
    tap += 16;
    if (tap >= 9) { tap -= 9; ++cin; }
    if (tap >= 9) { tap -= 9; ++cin; }
  };

  auto commit = [&]() {
    wait_asynccnt0();
    __syncthreads();
  };

  auto mma_tile = [&](int bu) {
#pragma unroll
    for (int ks = 0; ks < 4; ++ks) {
      const int kk = ks * 4 + ((lane >> 4) << 1);
      const int ml = wm * 16 + (lane & 15);
      v2f af;
      af.x = As[bu][ml][kk];
      af.y = As[bu][ml][kk + 1];
#pragma unroll
      for (int t = 0; t < 4; ++t) {
        const int nl = wn * 64 + t * 16 + (lane & 15);
        v2f bf;
        bf.x = Bs[bu][kk][nl];
        bf.y = Bs[bu][kk + 1][nl];
        acc[t] = wmma_f32(af, bf, acc[t]);
      }
    }
  };

  stage(0, 0);
  commit();
  int buf = 0;
  for (int k0 = 0; k0 < K; k0 += BK) {
    if (k0 + BK < K) stage(buf ^ 1, k0 + BK);
    mma_tile(buf);
    commit();
    buf ^= 1;
  }

  const int half = lane >> 4, col = lane & 15;
#pragma unroll
  for (int r = 0; r < 8; ++r) {
    const int m = m0 + wm * 16 + r + 8 * half;
    const float a = gs[m] * rsqrtf(gv[m] + 1e-5f);
    const float bb = gb[m] - a * gm[m];
#pragma unroll
    for (int t = 0; t < 4; ++t) {
      const int n = n0 + wn * 64 + t * 16 + col;
      Ob[(long long)m * 4096 + n] = fmaxf(fmaf(a, acc[t][r], bb), 0.0f);
    }
  }
}

// ===========================================================================
// Row softmax (PAM: G=K^T Q, so softmax(E,axis=1) == row-softmax of G).
// ===========================================================================
__global__ __launch_bounds__(256) void row_softmax_k(float* __restrict__ G,
                                                     int width) {
  __shared__ float red[256];
  float* row = G + (long long)blockIdx.x * width;
  const int tid = threadIdx.x;
  float m = -3.4e38f;
  for (int i = tid; i < width; i += 256) m = fmaxf(m, row[i]);
  red[tid] = m;
  __syncthreads();
  for (int s = 128; s > 0; s >>= 1) {
    if (tid < s) red[tid] = fmaxf(red[tid], red[tid + s]);
    __syncthreads();
  }
  m = red[0];
  __syncthreads();
  float sum = 0.0f;
  for (int i = tid; i < width; i += 256) {
    float e = expf(row[i] - m);
    row[i] = e;
    sum += e;
  }
  red[tid] = sum;
  __syncthreads();
  for (int s = 128; s > 0; s >>= 1) {
    if (tid < s) red[tid] += red[tid + s];
    __syncthreads();
  }
  const float inv = 1.0f / red[0];
  for (int i = tid; i < width; i += 256) row[i] *= inv;
}

// CAM: softmax(rowmax - E, axis=-1) == exp(rowmin - E)/sum (stable, exact).
__global__ __launch_bounds__(256) void cam_softmax_k(float* __restrict__ E) {
  __shared__ float red[256];
  float* row = E + (long long)blockIdx.x * 512;
  const int tid = threadIdx.x;
  float mn = 3.4e38f;
  for (int i = tid; i < 512; i += 256) mn = fminf(mn, row[i]);
  red[tid] = mn;
  __syncthreads();
  for (int s = 128; s > 0; s >>= 1) {
    if (tid < s) red[tid] = fminf(red[tid], red[tid + s]);
    __syncthreads();
  }
  mn = red[0];
  __syncthreads();
  float sum = 0.0f;
  for (int i = tid; i < 512; i += 256) {
    float e = expf(mn - row[i]);
    row[i] = e;
    sum += e;
  }
  red[tid] = sum;
  __syncthreads();
  for (int s = 128; s > 0; s >>= 1) {
    if (tid < s) red[tid] += red[tid + s];
    __syncthreads();
  }
  const float inv = 1.0f / red[0];
  for (int i = tid; i < 512; i += 256) row[i] *= inv;
}

__global__ __launch_bounds__(256) void add_inplace_k(float* __restrict__ a,
                                                     const float* __restrict__ b,
                                                     long long n) {
  long long i = (long long)blockIdx.x * 256 + threadIdx.x;
  if (i < n) a[i] += b[i];
}

// ===========================================================================
extern "C" void kernel_launch(void* const* d_in, const int* in_sizes, int n_in,
                              void* d_out, int out_size, void* d_ws,
                              size_t ws_size, hipStream_t stream) {
  (void)in_sizes; (void)n_in; (void)out_size; (void)ws_size;
  const float* x        = (const float*)d_in[0];
  const float* conv5a_w = (const float*)d_in[1];
  const float* bn5a_s = (const float*)d_in[2], *bn5a_b = (const float*)d_in[3];
  const float* bn5a_m = (const float*)d_in[4], *bn5a_v = (const float*)d_in[5];
  const float* conv5c_w = (const float*)d_in[6];
  const float* bn5c_s = (const float*)d_in[7], *bn5c_b = (const float*)d_in[8];
  const float* bn5c_m = (const float*)d_in[9], *bn5c_v = (const float*)d_in[10];
  const float* pam_q_w = (const float*)d_in[11], *pam_q_b = (const float*)d_in[12];
  const float* pam_k_w = (const float*)d_in[13], *pam_k_b = (const float*)d_in[14];
  const float* pam_v_w = (const float*)d_in[15], *pam_v_b = (const float*)d_in[16];
  const float* pam_gamma = (const float*)d_in[17];
  const float* cam_gamma = (const float*)d_in[18];
  const float* conv51_w = (const float*)d_in[19];
  const float* bn51_s = (const float*)d_in[20], *bn51_b = (const float*)d_in[21];
  const float* bn51_m = (const float*)d_in[22], *bn51_v = (const float*)d_in[23];
  const float* conv52_w = (const float*)d_in[24];
  const float* bn52_s = (const float*)d_in[25], *bn52_b = (const float*)d_in[26];
  const float* bn52_m = (const float*)d_in[27], *bn52_v = (const float*)d_in[28];
  const float* conv8_w = (const float*)d_in[29], *conv8_b = (const float*)d_in[30];
  float* out = (float*)d_out;

  const int B = 4, N = 4096, CI = 512;
  const long long F = (long long)CI * N;  // per-batch feature map (512x4096)

  // workspace layout (~200 MiB of fp32)
  float* ws = (float*)d_ws;
  float* buf0 = ws;                       // feat1, later feat2      (B*F)
  float* buf1 = buf0 + (long long)B * F;  // v, later sc_feat        (B*F)
  float* buf2 = buf1 + (long long)B * F;  // sa_feat, later sc_conv  (B*F)
  float* buf3 = buf2 + (long long)B * F;  // sa_conv -> feat_sum     (B*F)
  float* bufQ = buf3 + (long long)B * F;  // q  (B*64*4096)
  float* bufK = bufQ + (long long)B * 64 * N;  // k
  float* bufG = bufK + (long long)B * 64 * N;  // G (4096x4096 per batch) / camE

  dim3 blk(256);

  // 1) feat1 = bnrelu(conv3x3(x, conv5a))
  conv3x3_bnrelu_k<<<dim3(32, 8, B), blk, 0, stream>>>(
      x, conv5a_w, bn5a_s, bn5a_b, bn5a_m, bn5a_v, buf0, 2048);

  // 2-4) q,k (64xN), v (512xN): 1x1 convs == GEMM(W, feat1) + bias
  gemm_wmma_k<0, 0, 1, 0><<<dim3(32, 1, B), blk, 0, stream>>>(
      pam_q_w, buf0, bufQ, 64, N, CI, 0, F, (long long)64 * N, pam_q_b,
      nullptr, 0);
  gemm_wmma_k<0, 0, 1, 0><<<dim3(32, 1, B), blk, 0, stream>>>(
      pam_k_w, buf0, bufK, 64, N, CI, 0, F, (long long)64 * N, pam_k_b,
      nullptr, 0);
  gemm_wmma_k<0, 0, 1, 0><<<dim3(32, 8, B), blk, 0, stream>>>(
      pam_v_w, buf0, buf1, CI, N, CI, 0, F, F, pam_v_b, nullptr, 0);

  // 5) PAM per batch: G = K^T Q ; row-softmax ; sa_feat = g*(V x G) + feat1
  for (int b = 0; b < B; ++b) {
    const float* qb = bufQ + (long long)b * 64 * N;
    const float* kb = bufK + (long long)b * 64 * N;
    const float* vb = buf1 + (long long)b * F;
    const float* f1 = buf0 + (long long)b * F;
    float* saf = buf2 + (long long)b * F;
    gemm_wmma_k<1, 0, 0, 0><<<dim3(32, 64, 1), blk, 0, stream>>>(
        kb, qb, bufG, N, N, 64, 0, 0, 0, nullptr, nullptr, 0);
    row_softmax_k<<<dim3(N), blk, 0, stream>>>(bufG, N);
    gemm_wmma_k<0, 0, 2, 0><<<dim3(32, 8, 1), blk, 0, stream>>>(
        vb, bufG, saf, CI, N, N, 0, 0, 0, pam_gamma, f1, 0);
  }

  // 6) sa_conv = bnrelu(conv3x3(sa_feat, conv51))
  conv3x3_bnrelu_k<<<dim3(32, 8, B), blk, 0, stream>>>(
      buf2, conv51_w, bn51_s, bn51_b, bn51_m, bn51_v, buf3, 512);

  // 7) feat2 = bnrelu(conv3x3(x, conv5c))
  conv3x3_bnrelu_k<<<dim3(32, 8, B), blk, 0, stream>>>(
      x, conv5c_w, bn5c_s, bn5c_b, bn5c_m, bn5c_v, buf0, 2048);

  // 8) CAM energy: E = F F^T (512x512 per batch, NT GEMM)
  gemm_wmma_k<0, 1, 0, 0><<<dim3(4, 8, B), blk, 0, stream>>>(
      buf0, buf0, bufG, CI, CI, N, F, F, (long long)CI * CI, nullptr, nullptr,
      0);
  // 9) softmax(rowmax - E) rows
  cam_softmax_k<<<dim3(B * CI), blk, 0, stream>>>(bufG);
  // 10) sc_feat = g*(attn x F) + feat2
  gemm_wmma_k<0, 0, 2, 0><<<dim3(32, 8, B), blk, 0, stream>>>(
      bufG, buf0, buf1, CI, N, CI, (long long)CI * CI, F, F, cam_gamma, buf0,
      F);

  // 11) sc_conv = bnrelu(conv3x3(sc_feat, conv52))
  conv3x3_bnrelu_k<<<dim3(32, 8, B), blk, 0, stream>>>(
      buf1, conv52_w, bn52_s, bn52_b, bn52_m, bn52_v, buf2, 512);

  // 12) feat_sum = sa_conv + sc_conv (into buf3)
  add_inplace_k<<<dim3((int)(((long long)B * F + 255) / 256)), blk, 0,
                  stream>>>(buf3, buf2, (long long)B * F);

  // 13) out = conv1x1(feat_sum, conv8) + bias (M=19, masked VGPR staging)
  gemm_wmma_k<0, 0, 1, 1><<<dim3(32, 1, B), blk, 0, stream>>>(
      conv8_w, buf3, out, 19, N, CI, 0, F, (long long)19 * N, conv8_b, nullptr,
      0);
}